// TriangleScene_37709812859286
// MI455X (gfx1250) — compile-verified
//
#include <hip/hip_runtime.h>
#include <math.h>

#define EPS      1e-6f
#define HIT_TOL  1e-3f
#define MIN_LEN  1.2e-6f

typedef __attribute__((ext_vector_type(2))) float v2f;
typedef __attribute__((ext_vector_type(8))) float v8f;

struct F3 { float x, y, z; };
__device__ __forceinline__ F3 add3(F3 a, F3 b){ return {a.x+b.x, a.y+b.y, a.z+b.z}; }
__device__ __forceinline__ F3 sub3(F3 a, F3 b){ return {a.x-b.x, a.y-b.y, a.z-b.z}; }
__device__ __forceinline__ F3 scl3(F3 a, float s){ return {a.x*s, a.y*s, a.z*s}; }
__device__ __forceinline__ float dot3(F3 a, F3 b){ return a.x*b.x + a.y*b.y + a.z*b.z; }
__device__ __forceinline__ F3 cross3(F3 a, F3 b){
    return {a.y*b.z - a.z*b.y, a.z*b.x - a.x*b.z, a.x*b.y - a.y*b.x};
}
__device__ __forceinline__ F3 ld3(const float* p){ return {p[0], p[1], p[2]}; }
__device__ __forceinline__ void st3(float* p, F3 v){ p[0]=v.x; p[1]=v.y; p[2]=v.z; }

// Exact reference Moller-Trumbore for the 2 "inside" tests (cheap, 2 per item).
__device__ __forceinline__ bool mt_hit(F3 ro, F3 rd, F3 v0, F3 e1, F3 e2) {
    F3 h = cross3(rd, e2);
    float a = dot3(e1, h);
    bool cond_a = fabsf(a) < EPS;
    float f = cond_a ? 0.f : (1.f / a);
    F3 s = sub3(ro, v0);
    float u = f * dot3(s, h);
    F3 q = cross3(s, e1);
    float v = f * dot3(rd, q);
    float tt = f * dot3(e2, q);
    return (!cond_a) && (u >= 0.f) && (u <= 1.f) && (v >= 0.f) && ((u + v) <= 1.f) && (tt > EPS);
}

// ---------------- K0: per-triangle geometry + WMMA feature columns -----------
// triGeo[f]  : v0(3), e1(3), e2(3), n_normalized(3)       -> 12 floats
// triCols[c] : 12-float K-column; c = 4*f + role
//   role0 (a): [-fn, 0,0,0, 0,0,0, 0,0,0]
//   role1 (U): [v0 x e2, e2, 0,0,0, 0,0,0]
//   role2 (V): [e1 x v0, -e1, 0,0,0, 0,0,0]
//   role3 (T): [0,0,0, 0,0,0, fn, -d, 0,0]   (d = fn . v0)
__global__ void k_tri(const float* __restrict__ mv, const int* __restrict__ mt,
                      float* __restrict__ triGeo, float* __restrict__ triCols,
                      int NF, int NFpad)
{
    int f = blockIdx.x * blockDim.x + threadIdx.x;
    if (f >= NFpad) return;
    if (f >= NF) {                        // zero-fill padding columns -> never "blocked"
        for (int c = 0; c < 4; ++c)
            for (int k = 0; k < 12; ++k) triCols[(size_t)(f*4 + c)*12 + k] = 0.f;
        return;
    }
    int i0 = mt[f*3+0], i1 = mt[f*3+1], i2 = mt[f*3+2];
    F3 v0 = ld3(mv + (size_t)i0*3);
    F3 v1 = ld3(mv + (size_t)i1*3);
    F3 v2 = ld3(mv + (size_t)i2*3);
    F3 e1 = sub3(v1, v0);
    F3 e2 = sub3(v2, v0);
    F3 fn = cross3(e1, e2);
    float inv = 1.f / sqrtf(dot3(fn, fn));
    F3 n = scl3(fn, inv);

    float* g = triGeo + (size_t)f*12;
    st3(g+0, v0); st3(g+3, e1); st3(g+6, e2); st3(g+9, n);

    float d = dot3(fn, v0);
    F3 v0xe2 = cross3(v0, e2);
    F3 e1xv0 = cross3(e1, v0);

    float* c0 = triCols + (size_t)(f*4 + 0)*12;
    float* c1 = triCols + (size_t)(f*4 + 1)*12;
    float* c2 = triCols + (size_t)(f*4 + 2)*12;
    float* c3 = triCols + (size_t)(f*4 + 3)*12;
    #pragma unroll
    for (int k = 0; k < 12; ++k) { c0[k]=0.f; c1[k]=0.f; c2[k]=0.f; c3[k]=0.f; }
    c0[0]=-fn.x; c0[1]=-fn.y; c0[2]=-fn.z;
    c1[0]=v0xe2.x; c1[1]=v0xe2.y; c1[2]=v0xe2.z; c1[3]=e2.x; c1[4]=e2.y; c1[5]=e2.z;
    c2[0]=e1xv0.x; c2[1]=e1xv0.y; c2[2]=e1xv0.z; c2[3]=-e1.x; c2[4]=-e1.y; c2[5]=-e1.z;
    c3[6]=fn.x; c3[7]=fn.y; c3[8]=fn.z; c3[9]=-d;
}

// ---------------- K1: image method + per-item mask terms + ray features ------
__global__ void k_geom(const float* __restrict__ txv, const float* __restrict__ rxv,
                       const int* __restrict__ pc, const float* __restrict__ triGeo,
                       float* __restrict__ out_paths, float* __restrict__ out_obj,
                       float* __restrict__ rayFeat, float* __restrict__ partial,
                       int T, int R, int P)
{
    int item = blockIdx.x * blockDim.x + threadIdx.x;
    int NI = T * R * P;
    if (item >= NI) return;
    int p = item % P;
    int r = (item / P) % R;
    int t = item / (P * R);
    int f0 = pc[p*2 + 0];
    int f1 = pc[p*2 + 1];
    const float* g0 = triGeo + (size_t)f0*12;
    const float* g1 = triGeo + (size_t)f1*12;
    F3 mv0 = ld3(g0+0), e10 = ld3(g0+3), e20 = ld3(g0+6), n0 = ld3(g0+9);
    F3 mv1 = ld3(g1+0), e11 = ld3(g1+3), e21 = ld3(g1+6), n1 = ld3(g1+9);
    F3 tx = ld3(txv + (size_t)t*3);
    F3 rx = ld3(rxv + (size_t)r*3);

    // forward images
    F3 img0 = sub3(tx,   scl3(n0, 2.f * dot3(sub3(tx,   mv0), n0)));
    F3 img1 = sub3(img0, scl3(n1, 2.f * dot3(sub3(img0, mv1), n1)));
    // backward (reverse scan)
    F3 u1 = sub3(rx, img1);
    float un1 = dot3(u1, n1);
    float vn1 = dot3(sub3(img1, mv1), n1);
    float tt1 = (un1 == 0.f) ? 0.f : (-vn1 / un1);
    F3 p1 = add3(img1, scl3(u1, tt1));
    F3 u0 = sub3(p1, img0);
    float un0 = dot3(u0, n0);
    float vn0 = dot3(sub3(img0, mv0), n0);
    float tt0 = (un0 == 0.f) ? 0.f : (-vn0 / un0);
    F3 p0 = add3(img0, scl3(u0, tt0));

    // outputs: full_paths (T,R,P,4,3) and objects (T,R,P,4)
    float* fp = out_paths + (size_t)item*12;
    st3(fp+0, tx); st3(fp+3, p0); st3(fp+6, p1); st3(fp+9, rx);
    float* ob = out_obj + (size_t)item*4;
    ob[0]=(float)t; ob[1]=(float)f0; ob[2]=(float)f1; ob[3]=(float)r;

    F3 pts[4] = {tx, p0, p1, rx};
    F3 ro[3]  = {tx, p0, p1};
    F3 rd[3];
    #pragma unroll
    for (int s = 0; s < 3; ++s) rd[s] = sub3(pts[s+1], pts[s]);

    bool inside = mt_hit(tx, rd[0], mv0, e10, e20) && mt_hit(p0, rd[1], mv1, e11, e21);

    float dp0 = dot3(sub3(tx, mv0), n0);
    float dn0 = dot3(sub3(p1, mv0), n0);
    float dp1 = dot3(sub3(p0, mv1), n1);
    float dn1 = dot3(sub3(rx, mv1), n1);
    bool valid = (dp0*dn0 >= 0.f) && (dp1*dn1 >= 0.f);

    bool small = false;
    #pragma unroll
    for (int s = 0; s < 3; ++s) small = small || (dot3(rd[s], rd[s]) < MIN_LEN);

    partial[item] = (inside && valid && !small) ? 1.f : 0.f;

    // ray features: [rd(3), ro x rd(3), ro(3), 1, 0, 0]
    #pragma unroll
    for (int s = 0; s < 3; ++s) {
        float* X = rayFeat + ((size_t)item*3 + s)*12;
        F3 m = cross3(ro[s], rd[s]);
        X[0]=rd[s].x; X[1]=rd[s].y; X[2]=rd[s].z;
        X[3]=m.x;     X[4]=m.y;     X[5]=m.z;
        X[6]=ro[s].x; X[7]=ro[s].y; X[8]=ro[s].z;
        X[9]=1.f;     X[10]=0.f;    X[11]=0.f;
    }
}

// Branchless, division-free Moller-Trumbore acceptance from the 4 bilinear
// forms (a, U, V, T): multiply all ratio tests through by |a| with the sign
// of `a` folded in.  u=U/a, v=V/a, t=T/a.
__device__ __forceinline__ int blk_pred(float av, float Uv, float Vv, float Tv) {
    float sg = copysignf(1.f, av);
    float aa = fabsf(av);
    float Us = Uv * sg, Vs = Vv * sg, Ts = Tv * sg;
    return (int)(aa >= EPS) & (int)(Us >= 0.f) & (int)(Us <= aa) &
           (int)(Vs >= 0.f) & (int)((Us + Vs) <= aa) &
           (int)(Ts > EPS * aa) & (int)(Ts < (1.f - HIT_TOL) * aa);
}

// ---------------- K2: occlusion GEMM via V_WMMA_F32_16X16X4_F32 --------------
// Transposed mapping: M = triangle feature columns, N = rays.  One wave per
// ray tile (16 rays); loop over all column tiles (16 cols = 4 triangles).
// D layout (lane = N-col, VGPR = M-row) puts 2 complete triangles (4 roles
// each) in each lane's own registers -> no cross-lane ops in the inner loop.
__global__ void k_blocked(const float* __restrict__ rayFeat,
                          const float* __restrict__ triCols,
                          unsigned* __restrict__ blockedMask,
                          int numRayTiles, int numColTiles)
{
    int gtid = blockIdx.x * blockDim.x + threadIdx.x;
    int wave = gtid >> 5;
    int lane = threadIdx.x & 31;
    if (wave >= numRayTiles) return;         // wave-uniform: EXEC all-1s inside

    int half = lane >> 4;                    // 0 or 1
    int lmod = lane & 15;
    int kA   = half * 2;                     // lanes 0-15: K={0,1}; 16-31: K={2,3}

    // B fragment (rays), loaded once: B[k][n] = X[ray = wave*16+n][k]
    const float* xrow = rayFeat + (size_t)(wave*16 + lmod) * 12;
    v2f b0 = *(const v2f*)(xrow + 0 + kA);
    v2f b1 = *(const v2f*)(xrow + 4 + kA);
    v2f b2 = *(const v2f*)(xrow + 8 + kA);

    // A fragment (triangle columns), double-buffered across the tile loop.
    size_t colOff = (size_t)lmod * 12 + kA;
    v2f ta0 = *(const v2f*)(triCols + colOff + 0);
    v2f ta1 = *(const v2f*)(triCols + colOff + 4);
    v2f ta2 = *(const v2f*)(triCols + colOff + 8);

    int acc = 0;                             // per-lane: ray blocked by any of
                                             // this lane's triangle residues
    for (int mtile = 0; mtile < numColTiles; ++mtile) {
        int nxt = (mtile + 1 < numColTiles) ? (mtile + 1) : mtile;
        const float* yc = triCols + (size_t)nxt * 16 * 12 + colOff;
        v2f na0 = *(const v2f*)(yc + 0);
        v2f na1 = *(const v2f*)(yc + 4);
        v2f na2 = *(const v2f*)(yc + 8);

        v8f c = {0.f,0.f,0.f,0.f,0.f,0.f,0.f,0.f};
        c = __builtin_amdgcn_wmma_f32_16x16x4_f32(false, ta0, false, b0, (short)0, c, false, false);
        c = __builtin_amdgcn_wmma_f32_16x16x4_f32(false, ta1, false, b1, (short)0, c, false, false);
        c = __builtin_amdgcn_wmma_f32_16x16x4_f32(false, ta2, false, b2, (short)0, c, false, false);

        // Lane rows: half==0 -> M rows 0..7 (triangles 4*mtile+{0,1}),
        //            half==1 -> M rows 8..15 (triangles 4*mtile+{2,3}).
        acc |= blk_pred(c[0], c[1], c[2], c[3]);
        acc |= blk_pred(c[4], c[5], c[6], c[7]);

        ta0 = na0; ta1 = na1; ta2 = na2;
    }

    // Combine the two halves (same ray, different triangle residues) and pack
    // the 16 per-ray blocked bits of this tile.
    unsigned m32 = (unsigned)__ballot(acc);
    unsigned m16 = (m32 | (m32 >> 16)) & 0xffffu;
    if (lane == 0) blockedMask[wave] = m16;
}

// ---------------- K3: final mask combine -------------------------------------
__global__ void k_final(const float* __restrict__ partial,
                        const unsigned* __restrict__ blockedMask,
                        float* __restrict__ out_mask, int NI)
{
    int item = blockIdx.x * blockDim.x + threadIdx.x;
    if (item >= NI) return;
    bool blk = false;
    #pragma unroll
    for (int s = 0; s < 3; ++s) {
        int ray = item*3 + s;
        blk = blk || ((blockedMask[ray >> 4] >> (ray & 15)) & 1u);
    }
    out_mask[item] = (partial[item] != 0.f && !blk) ? 1.f : 0.f;
}

extern "C" void kernel_launch(void* const* d_in, const int* in_sizes, int n_in,
                              void* d_out, int out_size, void* d_ws, size_t ws_size,
                              hipStream_t stream)
{
    const float* mesh_v = (const float*)d_in[0];
    const int*   mesh_t = (const int*)d_in[1];
    const float* txv    = (const float*)d_in[2];
    const float* rxv    = (const float*)d_in[3];
    const int*   pc     = (const int*)d_in[4];

    int NF = in_sizes[1] / 3;
    int T  = in_sizes[2] / 3;
    int R  = in_sizes[3] / 3;
    const int ORDER = 2;
    int P  = in_sizes[4] / ORDER;
    int NI = T * R * P;
    int NR = NI * 3;

    int numRayTiles = (NR + 15) / 16;
    int NFpad       = ((NF + 3) / 4) * 4;      // columns padded to multiple of 16
    int numColTiles = (NFpad * 4) / 16;

    char* ws = (char*)d_ws;
    size_t off = 0;
    auto alloc = [&](size_t bytes) -> char* {
        char* q = ws + off;
        off = (off + bytes + 255) & ~(size_t)255;
        return q;
    };
    float*    triGeo      = (float*)   alloc((size_t)NF * 12 * sizeof(float));
    float*    triCols     = (float*)   alloc((size_t)NFpad * 4 * 12 * sizeof(float));
    float*    rayFeat     = (float*)   alloc((size_t)numRayTiles * 16 * 12 * sizeof(float));
    float*    partial     = (float*)   alloc((size_t)NI * sizeof(float));
    unsigned* blockedMask = (unsigned*)alloc((size_t)numRayTiles * sizeof(unsigned));
    (void)ws_size; (void)n_in; (void)out_size;

    float* out_paths = (float*)d_out;                 // NI * 12
    float* out_obj   = out_paths + (size_t)NI * 12;   // NI * 4
    float* out_mask  = out_obj   + (size_t)NI * 4;    // NI

    k_tri  <<<(NFpad + 127) / 128, 128, 0, stream>>>(mesh_v, mesh_t, triGeo, triCols, NF, NFpad);
    k_geom <<<(NI + 255) / 256, 256, 0, stream>>>(txv, rxv, pc, triGeo, out_paths, out_obj,
                                                  rayFeat, partial, T, R, P);
    int nthreads = numRayTiles * 32;
    k_blocked<<<(nthreads + 255) / 256, 256, 0, stream>>>(rayFeat, triCols, blockedMask,
                                                          numRayTiles, numColTiles);
    k_final<<<(NI + 255) / 256, 256, 0, stream>>>(partial, blockedMask, out_mask, NI);
}